// PreferenceCompletion_23570780520488
// MI455X (gfx1250) — compile-verified
//
#include <hip/hip_runtime.h>
#include <hip/hip_bf16.h>

typedef __attribute__((ext_vector_type(2))) float v2f;
typedef __attribute__((ext_vector_type(8))) float v8f;

#define WMMA4(a,b,c) __builtin_amdgcn_wmma_f32_16x16x4_f32(false,(a),false,(b),(short)0,(c),false,false)

namespace cfg {
constexpr int B=32, N=64, T=512, DIN=64, H=128, G=384, DZ=8;
constexpr int BN=B*N;
constexpr int MW=16;                    // sequences (M rows) per workgroup
constexpr int NWG=BN/MW;                // 128 workgroups
// LDS layout (floats). Even strides -> every v2f (b64) fragment load is 8B aligned.
constexpr int S_WIH=66, S_WHH=130, S_H=132, S_O=132;
constexpr int O_WIH=0;
constexpr int O_WHH=O_WIH + G*S_WIH;    // 25344
constexpr int O_H  =O_WHH + G*S_WHH;    // 75264
constexpr int O_OUT=O_H  + MW*S_H;      // 77376
constexpr int O_A  =O_OUT+ MW*S_O;      // 79488
constexpr int LDSF =O_A + MW;           // 79504 floats
constexpr int LDSB =LDSF*4;             // 318016 B  (< 320 KB CDNA5 WGP LDS)
// output layout: mean, logvar, eta, lam (each BN*DZ), alpha (BN*T)
constexpr int OFF_MEAN=0;
constexpr int OFF_LOGV=BN*DZ;
constexpr int OFF_ETA =2*BN*DZ;
constexpr int OFF_LAM =3*BN*DZ;
constexpr int OFF_ALPHA=4*BN*DZ;
}

__device__ __forceinline__ float sigm(float x){ return 1.f/(1.f+expf(-x)); }
__device__ __forceinline__ float softplusf(float x){ return (x>20.f)?x:log1pf(expf(x)); }

__global__ __launch_bounds__(128,1)
void gru_bayes_fused(const float* __restrict__ X,   const float* __restrict__ MSK,
                     const float* __restrict__ Wih, const float* __restrict__ bih,
                     const float* __restrict__ Whh, const float* __restrict__ bhh,
                     const float* __restrict__ Weta,const float* __restrict__ beta,
                     const float* __restrict__ Wlam,const float* __restrict__ blam,
                     const float* __restrict__ Wal, const float* __restrict__ bal,
                     float* __restrict__ out)
{
  using namespace cfg;
  extern __shared__ float lds[];
  const int tid=threadIdx.x;
  const int w   =tid>>5;        // wave id (wave32)
  const int lane=tid&31;
  const int half=lane>>4;       // WMMA lane-half
  const int lp  =lane&15;
  const int seqbase=blockIdx.x*MW;

  // ---- one-time: stage W_ih (384x64) and W_hh (384x128) into padded LDS ----
  for(int i=tid;i<G*(DIN/4);i+=128){
    int r=i>>4, c=(i&15)<<2;
    float4 v=*(const float4*)(Wih+(size_t)r*DIN+c);
    float* d=lds+O_WIH+r*S_WIH+c; d[0]=v.x; d[1]=v.y; d[2]=v.z; d[3]=v.w;
  }
  for(int i=tid;i<G*(H/4);i+=128){
    int r=i>>5, c=(i&31)<<2;
    float4 v=*(const float4*)(Whh+(size_t)r*H+c);
    float* d=lds+O_WHH+r*S_WHH+c; d[0]=v.x; d[1]=v.y; d[2]=v.z; d[3]=v.w;
  }
  for(int i=tid;i<MW*S_H;i+=128) lds[O_H+i]=0.f;   // h0 = 0
  __syncthreads();

  // ---- per-lane constants: wave w owns gate channels [32w, 32w+32) ----
  const int colR=32*w+lp;                 // channel col for tile-pair p via +16p
  float br[2],bzg[2],bnx[2],bnh[2];
  #pragma unroll
  for(int p=0;p<2;p++){
    int c=colR+16*p;
    br [p]=bih[c]      +bhh[c];
    bzg[p]=bih[H+c]    +bhh[H+c];
    bnx[p]=bih[2*H+c];
    bnh[p]=bhh[2*H+c];
  }
  const float *pIhR[2],*pIhZ[2],*pIhN[2],*pHhR[2],*pHhZ[2],*pHhN[2];
  #pragma unroll
  for(int p=0;p<2;p++){
    int c=colR+16*p;
    pIhR[p]=lds+O_WIH+(      c)*S_WIH+2*half;
    pIhZ[p]=lds+O_WIH+(H   +c)*S_WIH+2*half;
    pIhN[p]=lds+O_WIH+(2*H +c)*S_WIH+2*half;
    pHhR[p]=lds+O_WHH+(      c)*S_WHH+2*half;
    pHhZ[p]=lds+O_WHH+(H   +c)*S_WHH+2*half;
    pHhN[p]=lds+O_WHH+(2*H +c)*S_WHH+2*half;
  }
  const float* pAh  =lds+O_H  +lp*S_H+2*half;                 // h A-fragments
  const float* pAo  =lds+O_OUT+lp*S_O+2*half;                 // out A-fragments
  const float* xlane=X  +(size_t)(seqbase+lp)*T*DIN+2*half;   // x A-fragments
  const float* mlane=MSK+(size_t)(seqbase+8*half)*T;          // mask rows
  const float* pBpe =(lp<8)?(Weta+(size_t)lp*H+2*half):(Wlam+(size_t)(lp-8)*H+2*half);
  const float  bpe  =(lp<8)?beta[lp]:blam[lp-8];
  const float  ba0  =bal[0];

  float hreg[2][8];
  #pragma unroll
  for(int p=0;p<2;p++){
    #pragma unroll
    for(int v=0;v<8;v++) hreg[p][v]=0.f;
  }
  float el[8];                            // lanes lp<8 hold eta, lp>=8 hold lam
  #pragma unroll
  for(int v=0;v<8;v++) el[v]=(lp<8)?0.f:1.f;

  #pragma unroll 1
  for(int t=0;t<T;t++){
    const float* xrow=xlane+t*DIN;
    __builtin_prefetch(xrow+DIN,0,1);     // next timestep -> global_prefetch_b8

    v8f aR[2],aZ[2],aNX[2],aNH[2];
    #pragma unroll
    for(int p=0;p<2;p++){ aR[p]={}; aZ[p]={}; aNX[p]={}; aNH[p]={}; }

    // ---- xg = x_t @ W_ih^T : K=64 (16 k-steps of 4) ----
    #pragma unroll
    for(int ks=0;ks<16;ks++){
      v2f a=*(const v2f*)(xrow+4*ks);
      #pragma unroll
      for(int p=0;p<2;p++){
        aR [p]=WMMA4(a,*(const v2f*)(pIhR[p]+4*ks),aR [p]);
        aZ [p]=WMMA4(a,*(const v2f*)(pIhZ[p]+4*ks),aZ [p]);
        aNX[p]=WMMA4(a,*(const v2f*)(pIhN[p]+4*ks),aNX[p]);
      }
    }
    // ---- hg = h @ W_hh^T : K=128 (32 k-steps of 4) ----
    #pragma unroll
    for(int ks=0;ks<32;ks++){
      v2f a=*(const v2f*)(pAh+4*ks);
      #pragma unroll
      for(int p=0;p<2;p++){
        aR [p]=WMMA4(a,*(const v2f*)(pHhR[p]+4*ks),aR [p]);
        aZ [p]=WMMA4(a,*(const v2f*)(pHhZ[p]+4*ks),aZ [p]);
        aNH[p]=WMMA4(a,*(const v2f*)(pHhN[p]+4*ks),aNH[p]);
      }
    }

    // ---- gate nonlinearity: wave-local (each wave owns all 3 gates of its 32 ch) ----
    float mk[8], ovals[2][8];
    #pragma unroll
    for(int v=0;v<8;v++) mk[v]=mlane[v*T+t];
    #pragma unroll
    for(int p=0;p<2;p++){
      #pragma unroll
      for(int v=0;v<8;v++){
        float r   =sigm(aR[p][v]+br[p]);
        float z   =sigm(aZ[p][v]+bzg[p]);
        float n   =tanhf(aNX[p][v]+bnx[p]+r*(aNH[p][v]+bnh[p]));
        float hold=hreg[p][v];
        float hc  =(1.f-z)*n+z*hold;
        float o   =mk[v]*hc;
        hreg[p][v]=o+(1.f-mk[v])*hold;
        ovals[p][v]=o;
      }
    }
    __syncthreads();                      // all waves done reading old h
    #pragma unroll
    for(int p=0;p<2;p++){
      #pragma unroll
      for(int v=0;v<8;v++){
        int m=v+8*half;
        lds[O_H  +m*S_H+colR+16*p]=hreg[p][v];
        lds[O_OUT+m*S_O+colR+16*p]=ovals[p][v];
      }
    }
    __syncthreads();                      // new h / out visible

    // ---- projections: wave0 -> [eta|lam] tile, wave1 -> alpha tile ----
    v8f pacc={};
    if(w==0){
      #pragma unroll
      for(int ks=0;ks<32;ks++)
        pacc=WMMA4(*(const v2f*)(pAo+4*ks),*(const v2f*)(pBpe+4*ks),pacc);
    } else if(w==1){
      v2f bzv={0.f,0.f};
      #pragma unroll
      for(int ks=0;ks<32;ks++){
        v2f b=(lp==0)?*(const v2f*)(Wal+4*ks+2*half):bzv;
        pacc=WMMA4(*(const v2f*)(pAo+4*ks),b,pacc);
      }
      if(lp==0){
        #pragma unroll
        for(int v=0;v<8;v++){
          int m=v+8*half;
          float av=sigm(pacc[v]+ba0)*mk[v];
          lds[O_A+m]=av;
          out[OFF_ALPHA+(size_t)(seqbase+m)*T+t]=av;
        }
      }
    }
    __syncthreads();                      // alpha visible
    if(w==0){
      #pragma unroll
      for(int v=0;v<8;v++){
        int m=v+8*half;
        float a=lds[O_A+m];
        float x=pacc[v]+bpe;
        float d=(lp<8)?x:softplusf(x);
        el[v]=(1.f-a)*el[v]+a*d;          // eta/lam filter update
      }
    }
  }

  // ---- finalize: mean = eta/lam, logvar = -log(lam) ----
  if(w==0){
    #pragma unroll
    for(int v=0;v<8;v++){
      int m=v+8*half, seq=seqbase+m;
      float val=el[v];
      float lamv=__shfl(val,16*half+(lane&7)+8,32);   // pair eta-lane with lam-lane
      if(lp<8){
        out[OFF_ETA +seq*DZ+lp]=val;
        out[OFF_MEAN+seq*DZ+lp]=val/lamv;
        out[OFF_LOGV+seq*DZ+lp]=-logf(lamv);
      }else{
        out[OFF_LAM +seq*DZ+(lp-8)]=val;
      }
    }
  }
}

extern "C" void kernel_launch(void* const* d_in, const int* in_sizes, int n_in,
                              void* d_out, int out_size, void* d_ws, size_t ws_size,
                              hipStream_t stream){
  (void)in_sizes;(void)n_in;(void)out_size;(void)d_ws;(void)ws_size;
  const float* X   =(const float*)d_in[0];
  const float* MSK =(const float*)d_in[1];
  const float* Wih =(const float*)d_in[2];
  const float* bih =(const float*)d_in[3];
  const float* Whh =(const float*)d_in[4];
  const float* bhh =(const float*)d_in[5];
  const float* Weta=(const float*)d_in[6];
  const float* beta=(const float*)d_in[7];
  const float* Wlam=(const float*)d_in[8];
  const float* blam=(const float*)d_in[9];
  const float* Wal =(const float*)d_in[10];
  const float* bal =(const float*)d_in[11];
  (void)hipFuncSetAttribute(reinterpret_cast<const void*>(&gru_bayes_fused),
                            hipFuncAttributeMaxDynamicSharedMemorySize, cfg::LDSB);
  gru_bayes_fused<<<cfg::NWG,128,cfg::LDSB,stream>>>(X,MSK,Wih,bih,Whh,bhh,
                                                     Weta,beta,Wlam,blam,Wal,bal,
                                                     (float*)d_out);
}